// GraphNet_22686017257663
// MI455X (gfx1250) — compile-verified
//
#include <hip/hip_runtime.h>
#include <math.h>

typedef __attribute__((ext_vector_type(2))) float v2f;
typedef __attribute__((ext_vector_type(8))) float v8f;

#define NFEAT 128

// ---- init: deg starts at 1.0 (self-loop weight), agg at 0 ----
__global__ void k_init(float* __restrict__ deg, float* __restrict__ agg, int n) {
  int i = blockIdx.x * blockDim.x + threadIdx.x;
  if (i < n) { deg[i] = 1.0f; agg[i] = 0.0f; }
}

// ---- degree: scatter-add edge weights on target index ----
__global__ void k_deg(const int* __restrict__ col, const float* __restrict__ w,
                      float* __restrict__ deg, int E) {
  int e = blockIdx.x * blockDim.x + threadIdx.x;
  if (e < E) atomicAdd(&deg[col[e]], w[e]);
}

// ---- dinv = rsqrt(deg) ----
__global__ void k_dinv(const float* __restrict__ deg, float* __restrict__ dinv, int n) {
  int i = blockIdx.x * blockDim.x + threadIdx.x;
  if (i < n) {
    float d = deg[i];
    dinv[i] = d > 0.0f ? rsqrtf(d) : 0.0f;
  }
}

// ---- h = x @ W via V_WMMA_F32_16X16X4_F32; one wave32 per 16-row tile ----
// A (16x4 f32): lane m in [0,15] holds row m, K = {k0, k0+1}; lanes 16-31 hold
// the same rows with K = {k0+2, k0+3} (ISA 7.12.2, 32-bit A layout).
// B (4x16 f32): every column is the W chunk, so every column of D equals h.
// D (16x16 f32): VGPR i, lanes 0-15 -> (M=i, N=lane); lanes 16-31 -> (M=i+8).
// Hence lane 0 holds h[row0..row0+7], lane 16 holds h[row0+8..row0+15].
__global__ void k_gemv_wmma(const float* __restrict__ x, const float* __restrict__ W,
                            float* __restrict__ h, int n, int ntiles) {
  const int wave = (blockIdx.x * blockDim.x + threadIdx.x) >> 5;
  const int lane = threadIdx.x & 31;
  if (wave >= ntiles) return;            // wave-uniform: EXEC all-ones at WMMA
  const int row0 = wave * 16;
  const int m    = lane & 15;
  const int hi   = lane >> 4;            // 0: K=k0,k0+1   1: K=k0+2,k0+3
  int row = row0 + m;
  if (row >= n) row = n - 1;             // clamp (n is a multiple of 16 here)
  const float* __restrict__ xrow = x + (size_t)row * NFEAT;

  v8f acc = {};
#pragma unroll
  for (int k0 = 0; k0 < NFEAT; k0 += 4) {
    const int k = k0 + (hi << 1);
    v2f a  = *(const v2f*)(xrow + k);    // 8B, even-index aligned
    v2f bm = *(const v2f*)(W + k);       // broadcast W chunk to all 16 columns
    acc = __builtin_amdgcn_wmma_f32_16x16x4_f32(false, a, false, bm,
                                                (short)0, acc, false, false);
  }

  if ((lane & 15) == 0) {
    const int base = row0 + hi * 8;
#pragma unroll
    for (int i = 0; i < 8; ++i)
      if (base + i < n) h[base + i] = acc[i];
  }
}

// ---- edge scatter: agg[col] += dinv[row]*w*dinv[col]*h[row] ----
__global__ void k_scatter(const int* __restrict__ row, const int* __restrict__ col,
                          const float* __restrict__ w, const float* __restrict__ dinv,
                          const float* __restrict__ h, float* __restrict__ agg, int E) {
  int e = blockIdx.x * blockDim.x + threadIdx.x;
  if (e < E) {
    int r = row[e], c = col[e];
    float v = dinv[r] * w[e] * dinv[c] * h[r];
    atomicAdd(&agg[c], v);
  }
}

// ---- finalize: add self-loop term dinv^2 * h, bias, then mish ----
__global__ void k_final(const float* __restrict__ agg, const float* __restrict__ dinv,
                        const float* __restrict__ h, const float* __restrict__ b,
                        float* __restrict__ out, int n) {
  int i = blockIdx.x * blockDim.x + threadIdx.x;
  if (i < n) {
    float di = dinv[i];
    float v  = agg[i] + di * di * h[i] + b[0];
    float sp = (v > 20.0f) ? v : log1pf(expf(v));  // softplus, overflow-safe
    out[i]   = v * tanhf(sp);                      // mish
  }
}

extern "C" void kernel_launch(void* const* d_in, const int* in_sizes, int n_in,
                              void* d_out, int out_size, void* d_ws, size_t ws_size,
                              hipStream_t stream) {
  const float* x  = (const float*)d_in[0];   // [N,128]
  const int*   ei = (const int*)d_in[1];     // [2,E] flattened
  const float* ea = (const float*)d_in[2];   // [E]
  const float* W  = (const float*)d_in[3];   // [128]
  const float* b  = (const float*)d_in[4];   // [1]
  float* out = (float*)d_out;

  const int E = in_sizes[2];
  const int N = in_sizes[0] / NFEAT;
  const int* row = ei;       // source
  const int* col = ei + E;   // target

  float* ws   = (float*)d_ws;
  float* deg  = ws;
  float* dinv = ws + N;
  float* h    = ws + 2 * (size_t)N;
  float* agg  = ws + 3 * (size_t)N;

  const int T = 256;                       // 8 wave32s per block
  dim3 bn((N + T - 1) / T);
  dim3 be((E + T - 1) / T);

  k_init<<<bn, T, 0, stream>>>(deg, agg, N);
  k_deg <<<be, T, 0, stream>>>(col, ea, deg, E);
  k_dinv<<<bn, T, 0, stream>>>(deg, dinv, N);

  const int ntiles = (N + 15) / 16;
  const int wpb    = T / 32;
  dim3 bg((ntiles + wpb - 1) / wpb);
  k_gemv_wmma<<<bg, T, 0, stream>>>(x, W, h, N, ntiles);

  k_scatter<<<be, T, 0, stream>>>(row, col, ea, dinv, h, agg, E);
  k_final  <<<bn, T, 0, stream>>>(agg, dinv, h, b, out, N);
}